// Zip2ZipVocabParallelEmbedding_18056042512987
// MI455X (gfx1250) — compile-verified
//
#include <hip/hip_runtime.h>

// Zip2Zip vocab-parallel embedding gather, TDM edition.
//
// Roofline: 0 FLOPs, ~512 MB of traffic (8 KB row per token in + out) ->
// ~22 us at 23.3 TB/s. Pure data movement, so we use the CDNA5 Tensor Data
// Mover: one descriptor-driven DMA pulls the whole 8 KB row into LDS, a
// second pushes it to the output. One wave32 per token; all index math is
// scalar (block-uniform token id), so the vector pipe is essentially idle
// and the TDM engines do the copy.

typedef float f4 __attribute__((ext_vector_type(4)));
typedef unsigned int u32;
typedef unsigned long long u64;
typedef u32 u32x4 __attribute__((ext_vector_type(4)));
typedef int  i32x4 __attribute__((ext_vector_type(4)));
typedef int  i32x8 __attribute__((ext_vector_type(8)));

#define HDIM 2048   // hidden size (floats per row); fast path requires this

// ---- TDM wrappers (arity differs between toolchains; see CDNA5_HIP.md) ----
__device__ __forceinline__ void tdm_load(u32x4 g0, i32x8 g1) {
    i32x4 z4 = {0, 0, 0, 0};
#if __clang_major__ >= 23
    i32x8 z8 = {0, 0, 0, 0, 0, 0, 0, 0};
    __builtin_amdgcn_tensor_load_to_lds(g0, g1, z4, z4, z8, 0);
#else
    __builtin_amdgcn_tensor_load_to_lds(g0, g1, z4, z4, 0);
#endif
}

__device__ __forceinline__ void tdm_store(u32x4 g0, i32x8 g1) {
    i32x4 z4 = {0, 0, 0, 0};
#if __clang_major__ >= 23
    i32x8 z8 = {0, 0, 0, 0, 0, 0, 0, 0};
    __builtin_amdgcn_tensor_store_from_lds(g0, g1, z4, z4, z8, 0);
#else
    __builtin_amdgcn_tensor_store_from_lds(g0, g1, z4, z4, 0);
#endif
}

// D# group0: count=1 | lds_addr | global_addr(57b, bits 120:64) | type=2.
__device__ __forceinline__ u32x4 make_g0(u32 lds_addr, u64 ga) {
    u32x4 g0;
    g0[0] = 1u;                                        // count=1, user mode
    g0[1] = lds_addr;                                  // LDS byte address
    g0[2] = (u32)ga;                                   // global_addr[31:0]
    g0[3] = (u32)((ga >> 32) & 0x1FFFFFFu) | (2u << 30); // [56:32] | type=2
    return g0;
}

// D# group1 for a 1-row, 2048-elem, 4B/elem tile of a [1 x 2048] tensor.
__device__ __forceinline__ i32x8 make_g1() {
    i32x8 g1;
    g1[0] = 0x00020000;                 // wg_mask=0, data_size=2 (4B), no flags
    g1[1] = (HDIM & 0xFFFF) << 16;      // tensor_dim0[15:0] @ [31:16]
    g1[2] = ((HDIM >> 16) & 0xFFFF)     // tensor_dim0[31:16]
          | (1 << 16);                  // tensor_dim1 = 1
    g1[3] = (HDIM & 0xFFFF) << 16;      // tensor_dim1[31:16]=0 | tile_dim0=2048
    g1[4] = 1;                          // tile_dim1=1, tile_dim2=0
    g1[5] = HDIM;                       // tensor_dim0_stride[31:0] = 2048
    g1[6] = 0;                          // stride0[47:32]=0, stride1[15:0]=0
    g1[7] = 0;                          // stride1[47:16]=0
    return g1;
}

__global__ __launch_bounds__(32) void zip2zip_embed_tdm(
    const int* __restrict__ input_,
    const int* __restrict__ tok2batch,
    const int* __restrict__ pool_idx,
    const float* __restrict__ weight,
    const float* __restrict__ ebuf,
    float* __restrict__ out,
    int ivs, int max_hyper)
{
    __shared__ float row[HDIM];         // 8 KB staging tile

    const int token = blockIdx.x;
    const int id = input_[token];       // block-uniform -> s_load + scalar branch

    const float* src = nullptr;
    if (id < ivs) {
        src = weight + (size_t)id * HDIM;
    } else {
        const int hid = id - ivs;
        if (hid < max_hyper) {
            const int slot = pool_idx[tok2batch[token]];
            src = ebuf + ((size_t)slot * (size_t)max_hyper + (size_t)hid) * (size_t)HDIM;
        }
    }

    float* dst = out + (size_t)token * HDIM;

    if (!src) {
        // Out-of-range hyper id (reference guard): zero row, NT b128 stores.
        const f4 z = {0.f, 0.f, 0.f, 0.f};
        f4* d4 = reinterpret_cast<f4*>(dst);
        for (int i = threadIdx.x; i < HDIM / 4; i += 32)
            __builtin_nontemporal_store(z, d4 + i);
        return;
    }

    // Flat-LDS aperture: low 32 bits of the generic shared pointer are the
    // LDS byte address (ISA 10.2).
    const u32 lds_addr = (u32)(size_t)&row[0];
    const i32x8 g1 = make_g1();

    // DMA global row -> LDS, drain TENSORcnt, DMA LDS -> output row.
    tdm_load(make_g0(lds_addr, (u64)src), g1);
    __builtin_amdgcn_s_wait_tensorcnt(0);
    tdm_store(make_g0(lds_addr, (u64)dst), g1);
    // s_endpgm performs an implicit wait-idle, draining the pending store.
}

// ---- generic fallback (hidden != 2048): direct b128 copy, NT stores ----
__global__ __launch_bounds__(256) void zip2zip_embed_direct(
    const int* __restrict__ input_,
    const int* __restrict__ tok2batch,
    const int* __restrict__ pool_idx,
    const float* __restrict__ weight,
    const float* __restrict__ ebuf,
    float* __restrict__ out,
    int ivs, int max_hyper, int hidden)
{
    const int token = blockIdx.x;
    const int id = input_[token];
    const int nvec = hidden >> 2;
    f4* __restrict__ dst = reinterpret_cast<f4*>(out) + (size_t)token * nvec;

    const f4* src = nullptr;
    if (id < ivs) {
        src = reinterpret_cast<const f4*>(weight) + (size_t)id * nvec;
    } else {
        const int hid = id - ivs;
        if (hid < max_hyper) {
            const int slot = pool_idx[tok2batch[token]];
            src = reinterpret_cast<const f4*>(ebuf) +
                  ((size_t)slot * (size_t)max_hyper + (size_t)hid) * (size_t)nvec;
        }
    }

    if (src) {
        for (int i = threadIdx.x; i < nvec; i += 256)
            __builtin_nontemporal_store(src[i], dst + i);
    } else {
        const f4 z = {0.f, 0.f, 0.f, 0.f};
        for (int i = threadIdx.x; i < nvec; i += 256)
            __builtin_nontemporal_store(z, dst + i);
    }
}

extern "C" void kernel_launch(void* const* d_in, const int* in_sizes, int n_in,
                              void* d_out, int out_size, void* d_ws, size_t ws_size,
                              hipStream_t stream) {
    const int*   input_ = (const int*)  d_in[0];   // [N] token ids
    const int*   t2b    = (const int*)  d_in[1];   // [N] token -> batch
    const int*   pool   = (const int*)  d_in[2];   // [BATCH] batch -> pool slot
    const float* weight = (const float*)d_in[3];   // [IVS, H]
    const float* ebuf   = (const float*)d_in[4];   // [SLOTS, MAX_HYPER, H]
    float*       out    = (float*)      d_out;     // [N, H]

    const int n         = in_sizes[0];             // 32768
    const int hidden    = out_size / n;            // 2048
    const int ivs       = in_sizes[3] / hidden;    // 32000
    const int max_hyper = 1024;                    // reference MAX_HYPER

    if (hidden == HDIM) {
        zip2zip_embed_tdm<<<dim3((unsigned)n), dim3(32), 0, stream>>>(
            input_, t2b, pool, weight, ebuf, out, ivs, max_hyper);
    } else {
        zip2zip_embed_direct<<<dim3((unsigned)n), dim3(256), 0, stream>>>(
            input_, t2b, pool, weight, ebuf, out, ivs, max_hyper, hidden);
    }
}